// RUNG_percentile_gamma_59124519797084
// MI455X (gfx1250) — compile-verified
//
#include <hip/hip_runtime.h>
#include <math.h>

#define NN      4096
#define IN_DIM  512
#define HID     256
#define OUT_DIM 40
#define K_STEPS 10
#define E_MAX   (1 << 19)   // 524288 edge capacity (expected ~267K)

static constexpr float SCAD_Af  = 3.7f;
static constexpr float QQ       = 0.75f;
static constexpr float EPSV     = 1e-8f;
static constexpr float LAM_SKIP = (1.0f / 0.9f) - 1.0f;

typedef __attribute__((ext_vector_type(2))) float v2f;
typedef __attribute__((ext_vector_type(8))) float v8f;

// Radix-select state (lives in workspace)
struct QSel {
  unsigned prefix[2];     // bit prefixes of rank-lo / rank-hi values
  int      krem[2];       // remaining rank within current prefix
  int      n;             // number of edges
  float    frac;          // interpolation fraction
  unsigned hist[2][256];  // global histograms (one per target)
};

// ---------------------------------------------------------------------------
// GEMM1: H[N,HID] = relu(X[N,IN] @ W1[IN,HID] + b1)   (f32 WMMA 16x16x4)
// one wave per 32x16 output tile (2 M-tiles reuse each B fetch)
// ---------------------------------------------------------------------------
__global__ __launch_bounds__(32) void k_gemm1(const float* __restrict__ X,
                                              const float* __restrict__ W1,
                                              const float* __restrict__ b1,
                                              float* __restrict__ H) {
  const int lane = threadIdx.x;
  const int half = lane >> 4;        // 0: K=0,1 ; 1: K=2,3
  const int lm   = lane & 15;
  const int m0 = blockIdx.x * 32;
  const int n0 = blockIdx.y * 16;
  v8f acc0 = {0.f, 0.f, 0.f, 0.f, 0.f, 0.f, 0.f, 0.f};
  v8f acc1 = {0.f, 0.f, 0.f, 0.f, 0.f, 0.f, 0.f, 0.f};
  const float* xrow0 = X + (size_t)(m0 + lm) * IN_DIM + 2 * half;
  const float* xrow1 = xrow0 + (size_t)16 * IN_DIM;
  const float* bcol  = W1 + (size_t)(2 * half) * HID + n0 + lm;
#pragma unroll 8
  for (int k0 = 0; k0 < IN_DIM; k0 += 4) {
    float2 av0 = *(const float2*)(xrow0 + k0);
    float2 av1 = *(const float2*)(xrow1 + k0);
    v2f a0, a1, b;
    a0.x = av0.x; a0.y = av0.y;
    a1.x = av1.x; a1.y = av1.y;
    b.x = bcol[(size_t)k0 * HID];
    b.y = bcol[(size_t)k0 * HID + HID];
    acc0 = __builtin_amdgcn_wmma_f32_16x16x4_f32(false, a0, false, b,
                                                 (short)0, acc0, false, false);
    acc1 = __builtin_amdgcn_wmma_f32_16x16x4_f32(false, a1, false, b,
                                                 (short)0, acc1, false, false);
  }
  const float bias = b1[n0 + lm];
#pragma unroll
  for (int r = 0; r < 8; ++r) {
    const int m = m0 + r + 8 * half;           // C/D layout: M = r + 8*half
    float v0 = acc0[r] + bias;
    float v1 = acc1[r] + bias;
    H[(size_t)m * HID + n0 + lm]        = v0 > 0.f ? v0 : 0.f;
    H[(size_t)(m + 16) * HID + n0 + lm] = v1 > 0.f ? v1 : 0.f;
  }
}

// ---------------------------------------------------------------------------
// GEMM2: F0[N,40] = H[N,HID] @ W2[HID,40] + b2   (N padded to 48 = 3 tiles)
// ---------------------------------------------------------------------------
__global__ __launch_bounds__(32) void k_gemm2(const float* __restrict__ Hm,
                                              const float* __restrict__ W2,
                                              const float* __restrict__ b2,
                                              float* __restrict__ F0) {
  const int lane = threadIdx.x;
  const int half = lane >> 4;
  const int lm   = lane & 15;
  const int m0 = blockIdx.x * 32;
  const int n0 = blockIdx.y * 16;
  const int n  = n0 + lm;
  const bool nok = (n < OUT_DIM);
  v8f acc0 = {0.f, 0.f, 0.f, 0.f, 0.f, 0.f, 0.f, 0.f};
  v8f acc1 = {0.f, 0.f, 0.f, 0.f, 0.f, 0.f, 0.f, 0.f};
  const float* hrow0 = Hm + (size_t)(m0 + lm) * HID + 2 * half;
  const float* hrow1 = hrow0 + (size_t)16 * HID;
#pragma unroll 8
  for (int k0 = 0; k0 < HID; k0 += 4) {
    float2 av0 = *(const float2*)(hrow0 + k0);
    float2 av1 = *(const float2*)(hrow1 + k0);
    v2f a0, a1, b;
    a0.x = av0.x; a0.y = av0.y;
    a1.x = av1.x; a1.y = av1.y;
    const int kb = k0 + 2 * half;
    b.x = nok ? W2[(size_t)kb * OUT_DIM + n] : 0.f;
    b.y = nok ? W2[(size_t)(kb + 1) * OUT_DIM + n] : 0.f;
    acc0 = __builtin_amdgcn_wmma_f32_16x16x4_f32(false, a0, false, b,
                                                 (short)0, acc0, false, false);
    acc1 = __builtin_amdgcn_wmma_f32_16x16x4_f32(false, a1, false, b,
                                                 (short)0, acc1, false, false);
  }
  if (nok) {
    const float bias = b2[n];
#pragma unroll
    for (int r = 0; r < 8; ++r) {
      const int m = m0 + r + 8 * half;
      F0[(size_t)m * OUT_DIM + n]        = acc0[r] + bias;
      F0[(size_t)(m + 16) * OUT_DIM + n] = acc1[r] + bias;
    }
  }
}

// ---------------------------------------------------------------------------
// Per-row degree, D = deg+1, invDsq = 1/sqrt(D).  One block per row.
// ---------------------------------------------------------------------------
__global__ __launch_bounds__(256) void k_degree(const float* __restrict__ A,
                                                int* __restrict__ deg,
                                                float* __restrict__ Dv,
                                                float* __restrict__ invDsq) {
  __shared__ int sh[256];
  const int i = blockIdx.x;
  int c = 0;
  for (int j = threadIdx.x; j < NN; j += 256) {
    __builtin_prefetch(&A[(size_t)i * NN + j + 2048], 0, 0);
    c += (A[(size_t)i * NN + j] != 0.f) ? 1 : 0;
  }
  sh[threadIdx.x] = c;
  __syncthreads();
  for (int o = 128; o > 0; o >>= 1) {
    if (threadIdx.x < o) sh[threadIdx.x] += sh[threadIdx.x + o];
    __syncthreads();
  }
  if (threadIdx.x == 0) {
    const int d = sh[0];
    deg[i] = d;
    const float Df = (float)d + 1.0f;   // self loop
    Dv[i] = Df;
    invDsq[i] = rsqrtf(Df);
  }
}

// ---------------------------------------------------------------------------
// Exclusive scan of 4096 degrees -> rowptr[0..N], Ecnt.  Single block, 1024 thr.
// ---------------------------------------------------------------------------
__global__ __launch_bounds__(1024) void k_scan(const int* __restrict__ deg,
                                               int* __restrict__ rowptr,
                                               int* __restrict__ Ecnt) {
  __shared__ int s[1024];
  const int t = threadIdx.x;
  const int b = t * 4;
  const int a0 = deg[b], a1 = deg[b + 1], a2 = deg[b + 2], a3 = deg[b + 3];
  const int sum = a0 + a1 + a2 + a3;
  s[t] = sum;
  __syncthreads();
  for (int o = 1; o < 1024; o <<= 1) {
    int v = (t >= o) ? s[t - o] : 0;
    __syncthreads();
    s[t] += v;
    __syncthreads();
  }
  const int excl = s[t] - sum;
  rowptr[b]     = excl;
  rowptr[b + 1] = excl + a0;
  rowptr[b + 2] = excl + a0 + a1;
  rowptr[b + 3] = excl + a0 + a1 + a2;
  if (t == 1023) {
    rowptr[NN] = excl + sum;
    Ecnt[0]    = excl + sum;
  }
}

// ---------------------------------------------------------------------------
// CSR fill via deterministic wave-ordered ballot compaction. One wave per row.
// ---------------------------------------------------------------------------
__global__ __launch_bounds__(32) void k_fill(const float* __restrict__ A,
                                             const int* __restrict__ rowptr,
                                             const float* __restrict__ invDsq,
                                             int* __restrict__ row_i,
                                             int* __restrict__ col_i,
                                             float* __restrict__ atil) {
  const int i = blockIdx.x;
  const int lane = threadIdx.x;
  int base = rowptr[i];
  const float di = invDsq[i];
  for (int j0 = 0; j0 < NN; j0 += 32) {
    const int j = j0 + lane;
    __builtin_prefetch(&A[(size_t)i * NN + j + 512], 0, 0);
    const bool p = (A[(size_t)i * NN + j] != 0.f);
    const unsigned m = (unsigned)__ballot(p);
    const int pre = __popc(m & ((1u << lane) - 1u));
    if (p) {
      const int o = base + pre;
      row_i[o] = i;
      col_i[o] = j;
      atil[o]  = di * invDsq[j];
    }
    base += __popc(m);
  }
}

// ---------------------------------------------------------------------------
// Fn = F * invDsq (row-wise)
// ---------------------------------------------------------------------------
__global__ __launch_bounds__(256) void k_scale(const float* __restrict__ F,
                                               const float* __restrict__ invDsq,
                                               float* __restrict__ Fn) {
  const int idx = blockIdx.x * 256 + threadIdx.x;
  if (idx < NN * OUT_DIM) Fn[idx] = F[idx] * invDsq[idx / OUT_DIM];
}

// ---------------------------------------------------------------------------
// y_e = || Fn_i - Fn_j ||  per edge (float4-vectorized, 40 = 10x4)
// ---------------------------------------------------------------------------
__global__ __launch_bounds__(256) void k_edge_y(const float* __restrict__ Fn,
                                                const int* __restrict__ row_i,
                                                const int* __restrict__ col_i,
                                                const int* __restrict__ Ecnt,
                                                float* __restrict__ ye) {
  const int e = blockIdx.x * 256 + threadIdx.x;
  if (e >= Ecnt[0]) return;
  const float* fi = Fn + (size_t)row_i[e] * OUT_DIM;
  const float* fj = Fn + (size_t)col_i[e] * OUT_DIM;
  float s = 0.f;
#pragma unroll
  for (int c = 0; c < OUT_DIM; c += 4) {
    const float4 a = *(const float4*)(fi + c);
    const float4 b = *(const float4*)(fj + c);
    const float d0 = a.x - b.x, d1 = a.y - b.y, d2 = a.z - b.z, d3 = a.w - b.w;
    s += d0 * d0 + d1 * d1 + d2 * d2 + d3 * d3;
  }
  ye[e] = sqrtf(s);
}

// ---------------------------------------------------------------------------
// Grid-wide radix select, q=0.75 quantile with linear interpolation.
// k_qinit:  set up ranks lo / lo+1, zero global histograms.
// k_qhist:  one full pass over edge y values; LDS histograms for BOTH target
//           prefixes, flushed with integer global atomics (deterministic).
// k_qpick:  advance prefix/rank state by one byte; on last pass emit lam.
// ---------------------------------------------------------------------------
__global__ __launch_bounds__(256) void k_qinit(const int* __restrict__ Ecnt,
                                               QSel* __restrict__ st) {
  const int t = threadIdx.x;
  if (t == 0) {
    const int n = Ecnt[0];
    const float pos = QQ * (float)(n - 1);
    const int lo = (int)floorf(pos);
    const int hi = min(lo + 1, n - 1);
    st->prefix[0] = 0u;
    st->prefix[1] = 0u;
    st->krem[0] = lo;
    st->krem[1] = hi;
    st->n = n;
    st->frac = pos - (float)lo;
  }
  st->hist[0][t] = 0u;
  st->hist[1][t] = 0u;
}

__global__ __launch_bounds__(256) void k_qhist(const float* __restrict__ ye,
                                               QSel* __restrict__ st, int p) {
  __shared__ unsigned h0[256], h1[256];
  const int t = threadIdx.x;
  h0[t] = 0u;
  h1[t] = 0u;
  __syncthreads();
  const unsigned pre0 = st->prefix[0];
  const unsigned pre1 = st->prefix[1];
  const unsigned himask = (p == 3) ? 0u : (0xFFFFFFFFu << ((p + 1) * 8));
  const int n = st->n;
  for (int e = blockIdx.x * 256 + t; e < n; e += 64 * 256) {
    const unsigned u = __float_as_uint(ye[e]);   // y >= 0: bits order as u32
    const unsigned b = (u >> (p * 8)) & 255u;
    const unsigned hb = u & himask;
    if (hb == pre0) atomicAdd(&h0[b], 1u);
    if (hb == pre1) atomicAdd(&h1[b], 1u);
  }
  __syncthreads();
  if (h0[t]) atomicAdd(&st->hist[0][t], h0[t]);
  if (h1[t]) atomicAdd(&st->hist[1][t], h1[t]);
}

__global__ __launch_bounds__(256) void k_qpick(QSel* __restrict__ st, int p,
                                               float* __restrict__ lam) {
  const int t = threadIdx.x;
  if (t == 0) {
#pragma unroll
    for (int tgt = 0; tgt < 2; ++tgt) {
      const int krem = st->krem[tgt];
      unsigned cum = 0;
      int b = 0;
      for (; b < 255; ++b) {
        const unsigned c = st->hist[tgt][b];
        if (cum + c > (unsigned)krem) break;
        cum += c;
      }
      st->prefix[tgt] |= ((unsigned)b << (p * 8));
      st->krem[tgt] = krem - (int)cum;
    }
    if (p == 0) {
      const float v0 = __uint_as_float(st->prefix[0]);
      const float v1 = __uint_as_float(st->prefix[1]);
      const float g = v0 * (1.f - st->frac) + v1 * st->frac;
      const float gamma = (st->n > 0) ? fmaxf(g, EPSV) : SCAD_Af;
      lam[0] = gamma / SCAD_Af;
    }
  }
  __syncthreads();          // ensure thread 0 consumed hists before zeroing
  st->hist[0][t] = 0u;
  st->hist[1][t] = 0u;
}

// ---------------------------------------------------------------------------
// SCAD edge weights:  w, and w * atilde
// ---------------------------------------------------------------------------
__global__ __launch_bounds__(256) void k_edge_w(const float* __restrict__ ye,
                                                const float* __restrict__ atil,
                                                const float* __restrict__ lam,
                                                const int* __restrict__ Ecnt,
                                                float* __restrict__ we,
                                                float* __restrict__ wae) {
  const int e = blockIdx.x * 256 + threadIdx.x;
  if (e >= Ecnt[0]) return;
  const float l = lam[0];
  const float y = ye[e];
  float w;
  if (y <= l)                w = 1.f;
  else if (y <= SCAD_Af * l) w = (SCAD_Af * l - y) / ((SCAD_Af - 1.f) * y);
  else                       w = 0.f;
  if (w != w) w = 1.f;   // NaN -> 1 (reference)
  we[e]  = w;
  wae[e] = w * atil[e];
}

// ---------------------------------------------------------------------------
// Row update: Fdst_i = (sum_j wa_ij F_j + lam_skip F0_i) / (sum_j w_ij / D_i + lam_skip)
// One block per row, thread c owns channel c; fixed CSR order => deterministic.
// ---------------------------------------------------------------------------
__global__ __launch_bounds__(64) void k_rowupdate(const float* __restrict__ Fsrc,
                                                  const float* __restrict__ F0,
                                                  const float* __restrict__ we,
                                                  const float* __restrict__ wae,
                                                  const int* __restrict__ rowptr,
                                                  const int* __restrict__ col_i,
                                                  const float* __restrict__ Dv,
                                                  float* __restrict__ Fdst) {
  const int i = blockIdx.x;
  const int c = threadIdx.x;
  if (c >= OUT_DIM) return;
  const int e0 = rowptr[i], e1 = rowptr[i + 1];
  float acc = 0.f, s = 0.f;
  for (int e = e0; e < e1; ++e) {
    s   += we[e];
    acc += wae[e] * Fsrc[(size_t)col_i[e] * OUT_DIM + c];
  }
  const float Qh = s / Dv[i] + LAM_SKIP;
  Fdst[(size_t)i * OUT_DIM + c] = (acc + LAM_SKIP * F0[(size_t)i * OUT_DIM + c]) / Qh;
}

// ---------------------------------------------------------------------------
extern "C" void kernel_launch(void* const* d_in, const int* in_sizes, int n_in,
                              void* d_out, int out_size, void* d_ws, size_t ws_size,
                              hipStream_t stream) {
  const float* A  = (const float*)d_in[0];
  const float* X  = (const float*)d_in[1];
  const float* W1 = (const float*)d_in[2];
  const float* b1 = (const float*)d_in[3];
  const float* W2 = (const float*)d_in[4];
  const float* b2 = (const float*)d_in[5];
  float* out = (float*)d_out;

  char* base = (char*)d_ws;
  size_t off = 0;
  auto alloc = [&](size_t bytes) -> void* {
    void* p = base + off;
    off += (bytes + 255) & ~(size_t)255;
    return p;
  };
  float* H      = (float*)alloc((size_t)NN * HID * 4);
  float* F0     = (float*)alloc((size_t)NN * OUT_DIM * 4);
  float* Fa     = (float*)alloc((size_t)NN * OUT_DIM * 4);
  float* Fb     = (float*)alloc((size_t)NN * OUT_DIM * 4);
  float* Fn     = (float*)alloc((size_t)NN * OUT_DIM * 4);
  float* Dv     = (float*)alloc((size_t)NN * 4);
  float* invDsq = (float*)alloc((size_t)NN * 4);
  int*   deg    = (int*)alloc((size_t)NN * 4);
  int*   rowptr = (int*)alloc((size_t)(NN + 1) * 4);
  int*   Ecnt   = (int*)alloc(256);
  float* lam    = (float*)alloc(256);
  QSel*  qs     = (QSel*)alloc(sizeof(QSel));
  int*   row_i  = (int*)alloc((size_t)E_MAX * 4);
  int*   col_i  = (int*)alloc((size_t)E_MAX * 4);
  float* atil   = (float*)alloc((size_t)E_MAX * 4);
  float* ye     = (float*)alloc((size_t)E_MAX * 4);
  float* we     = (float*)alloc((size_t)E_MAX * 4);
  float* wae    = (float*)alloc((size_t)E_MAX * 4);
  (void)ws_size; (void)in_sizes; (void)n_in; (void)out_size;

  // MLP embedding (WMMA f32, 32x16 tile per wave)
  k_gemm1<<<dim3(NN / 32, HID / 16), 32, 0, stream>>>(X, W1, b1, H);
  k_gemm2<<<dim3(NN / 32, 3), 32, 0, stream>>>(H, W2, b2, F0);

  // Graph preprocessing (one-time per launch, deterministic)
  k_degree<<<NN, 256, 0, stream>>>(A, deg, Dv, invDsq);
  k_scan<<<1, 1024, 0, stream>>>(deg, rowptr, Ecnt);
  k_fill<<<NN, 32, 0, stream>>>(A, rowptr, invDsq, row_i, col_i, atil);

  // K_STEPS robust propagation iterations (sparse, edge-restricted)
  const int eblocks = E_MAX / 256;
  const float* src = F0;
  for (int it = 0; it < K_STEPS; ++it) {
    float* dst = (it == K_STEPS - 1) ? out : ((it & 1) ? Fb : Fa);
    k_scale<<<(NN * OUT_DIM + 255) / 256, 256, 0, stream>>>(src, invDsq, Fn);
    k_edge_y<<<eblocks, 256, 0, stream>>>(Fn, row_i, col_i, Ecnt, ye);
    k_qinit<<<1, 256, 0, stream>>>(Ecnt, qs);
    for (int p = 3; p >= 0; --p) {
      k_qhist<<<64, 256, 0, stream>>>(ye, qs, p);
      k_qpick<<<1, 256, 0, stream>>>(qs, p, lam);
    }
    k_edge_w<<<eblocks, 256, 0, stream>>>(ye, atil, lam, Ecnt, we, wae);
    k_rowupdate<<<NN, 64, 0, stream>>>(src, F0, we, wae, rowptr, col_i, Dv, dst);
    src = dst;
  }
}